// FilterDetections_66099546685469
// MI455X (gfx1250) — compile-verified
//
#include <hip/hip_runtime.h>
#include <stdint.h>

// ---------------------------------------------------------------------------
// FilterDetections for MI455X (gfx1250).
// Pipeline:
//   k_init    : zero histograms / counters in ws
//   k_hist    : TDM-streamed scan of classification -> 4096-bucket histogram
//               (tensor_load_to_lds double-buffered, s_wait_tensorcnt)
//   k_cutoff  : find cutoff bucket so that suffix count >= 300
//   k_collect : second scan, atomic-append candidate keys (score|~idx) >= cutoff
//   k_sort    : per-batch 4096-key bitonic sort (desc), keep top 300
//   k_gather  : gather boxes/loc/tr/rot + |conf| reduction, write outputs
// Outputs are written as float (labels/idx stored as float values, -1 invalid),
// concatenated in reference return order.
// Roofline: dominant cost is 2 passes over classification (2 x 48 MB) ->
// ~4.1 us at 23.3 TB/s; gathers are ~3.5 MB; sort/histogram are noise.
// ---------------------------------------------------------------------------

typedef unsigned int v4u __attribute__((ext_vector_type(4)));
typedef int          v8i __attribute__((ext_vector_type(8)));
typedef int          v4i __attribute__((ext_vector_type(4)));

static constexpr int B = 4, N = 100000, C = 30, DCONF = 15;
static constexpr int FLAT = N * C;              // 3,000,000 per batch
static constexpr int MAXDET = 300;
static constexpr int NBUCK = 4096;              // histogram buckets over (0.5, 1.0)
static constexpr int CAP = 4096;                // candidate capacity per batch
static constexpr int CHUNK = 4096;              // floats per TDM chunk (16 KB)

// output offsets (floats), reference return order
static constexpr long OFF_B3     = 0;                       // B*300*16
static constexpr long OFF_LOC    = 19200;                   // B*300*2
static constexpr long OFF_SCORES = 21600;                   // B*300
static constexpr long OFF_LABELS = 22800;                   // B*300
static constexpr long OFF_TR     = 24000;                   // B*300*C*3
static constexpr long OFF_ROT    = 132000;                  // B*300*C*6
static constexpr long OFF_CONF   = 348000;                  // B*300*C
static constexpr long OFF_IDX    = 384000;                  // B*300

// ws layout (bytes)
static constexpr size_t WS_HIST   = 0;                       // B*NBUCK*4 = 65536
static constexpr size_t WS_COUNT  = 65536;                   // B*4
static constexpr size_t WS_CUTOFF = 65552;                   // B*4
static constexpr size_t WS_CAND   = 65568;                   // B*CAP*8 = 131072
static constexpr size_t WS_SEL    = 196640;                  // B*300*8 = 9600

__device__ __forceinline__ int bucket_of(float s) {
    int bk = (int)((s - 0.5f) * 8192.0f);          // 4096 buckets over width 0.5
    bk = bk < 0 ? 0 : (bk > NBUCK - 1 ? NBUCK - 1 : bk);
    return bk;
}

// Issue one TDM load: `elems` contiguous f32 from gaddr -> LDS byte addr.
// D# per CDNA5 ISA ch.8: 2D tensor, dim0 = tile_dim0 = elems, dim1 = 1.
__device__ __forceinline__ void tdm_load_f32(uint64_t gaddr, uint32_t elems,
                                             uint32_t lds_addr) {
    v4u g0;
    g0[0] = 1u;                                              // count=1, user mode
    g0[1] = lds_addr;                                        // lds_addr [63:32]
    g0[2] = (uint32_t)gaddr;                                 // global_addr lo
    g0[3] = (uint32_t)((gaddr >> 32) & 0x1FFFFFFu) | (2u << 30); // addr hi | type=2
    v8i g1;
    g1[0] = (int)(2u << 16);                                 // data_size=2 (4 bytes)
    g1[1] = (int)((elems & 0xFFFFu) << 16);                  // tensor_dim0[15:0]
    g1[2] = (int)((elems >> 16) | (1u << 16));               // dim0 hi | tensor_dim1=1
    g1[3] = (int)((elems & 0xFFFFu) << 16);                  // tile_dim0 (<=4096)
    g1[4] = 1;                                               // tile_dim1=1, tile_dim2=0
    g1[5] = (int)elems;                                      // tensor_dim0_stride lo
    g1[6] = 0;                                               // stride hi | dim1_stride lo
    g1[7] = 0;
    v4i g2 = {0, 0, 0, 0};
    v4i g3 = {0, 0, 0, 0};
    v8i g4 = {0, 0, 0, 0, 0, 0, 0, 0};                       // extra group (clang-23 form)
    __builtin_amdgcn_tensor_load_to_lds(g0, g1, g2, g3, g4, 0);
}

__global__ void k_init(uint32_t* hist, uint32_t* counts) {
    int total = B * NBUCK + B;
    for (int i = blockIdx.x * blockDim.x + threadIdx.x; i < total;
         i += gridDim.x * blockDim.x) {
        if (i < B * NBUCK) hist[i] = 0u;
        else counts[i - B * NBUCK] = 0u;
    }
}

// ---- Pass 1: histogram of scores > 0.5, classification streamed via TDM ----
__global__ void k_hist(const float* __restrict__ cls, uint32_t* __restrict__ hist,
                       int chunksPerBlock, int blocksPerBatch) {
    __shared__ float buf[2][CHUNK];        // 32 KB double buffer
    __shared__ uint32_t lh[NBUCK];         // 16 KB local histogram

    const int b = blockIdx.x / blocksPerBatch;
    const int blk = blockIdx.x % blocksPerBatch;
    const long base = (long)b * FLAT;

    for (int i = threadIdx.x; i < NBUCK; i += blockDim.x) lh[i] = 0u;
    __syncthreads();

    const int totalChunks = (FLAT + CHUNK - 1) / CHUNK;
    const int firstChunk = blk * chunksPerBlock;
    int nC = totalChunks - firstChunk;
    if (nC > chunksPerBlock) nC = chunksPerBlock;

    const bool issuer = (threadIdx.x < 32);   // wave 0 drives the TDM

    if (nC > 0) {
        // elems in chunk c (relative to firstChunk)
        auto elemsOf = [&](int c) -> uint32_t {
            int s = (firstChunk + c) * CHUNK;
            int e = FLAT - s;
            return (uint32_t)(e > CHUNK ? CHUNK : e);
        };
        if (issuer) {
            uint64_t ga = (uint64_t)(uintptr_t)(cls + base + (long)firstChunk * CHUNK);
            tdm_load_f32(ga, elemsOf(0), (uint32_t)(uintptr_t)&buf[0][0]);
        }
        for (int c = 0; c < nC; ++c) {
            if (issuer) {
                if (c + 1 < nC) {
                    uint64_t ga = (uint64_t)(uintptr_t)
                        (cls + base + (long)(firstChunk + c + 1) * CHUNK);
                    tdm_load_f32(ga, elemsOf(c + 1),
                                 (uint32_t)(uintptr_t)&buf[(c + 1) & 1][0]);
                    __builtin_amdgcn_s_wait_tensorcnt(1);   // chunk c landed
                } else {
                    __builtin_amdgcn_s_wait_tensorcnt(0);
                }
            }
            __syncthreads();
            const int e = (int)elemsOf(c);
            const float* p = buf[c & 1];
            for (int i = threadIdx.x; i < e; i += blockDim.x) {
                float s = p[i];
                if (s > 0.5f) atomicAdd(&lh[bucket_of(s)], 1u);
            }
            __syncthreads();   // buffer reusable next iteration
        }
    }
    __syncthreads();
    for (int i = threadIdx.x; i < NBUCK; i += blockDim.x) {
        uint32_t v = lh[i];
        if (v) atomicAdd(&hist[b * NBUCK + i], v);
    }
}

// ---- Find cutoff bucket: largest t with suffix_sum(t) >= MAXDET ----
__global__ void k_cutoff(const uint32_t* __restrict__ hist,
                         uint32_t* __restrict__ cutoff) {
    const int b = blockIdx.x;
    if (threadIdx.x == 0) {
        const uint32_t* h = hist + (long)b * NBUCK;
        uint32_t sum = 0;
        int cb = 0;
        for (int i = NBUCK - 1; i >= 0; --i) {
            sum += h[i];
            if (sum >= (uint32_t)MAXDET) { cb = i; break; }
        }
        cutoff[b] = (uint32_t)cb;
    }
}

// ---- Pass 2: collect candidate keys (score_bits<<32 | ~flat_idx) ----
__global__ void k_collect(const float* __restrict__ cls,
                          const uint32_t* __restrict__ cutoff,
                          uint32_t* __restrict__ counts,
                          unsigned long long* __restrict__ cand) {
    const long total4 = (long)B * (FLAT / 4);     // FLAT % 4 == 0
    for (long i4 = (long)blockIdx.x * blockDim.x + threadIdx.x; i4 < total4;
         i4 += (long)gridDim.x * blockDim.x) {
        const long i = i4 * 4;
        const int b = (int)(i / FLAT);
        const int off = (int)(i - (long)b * FLAT);
        const float4 v = *(const float4*)(cls + i);
        const int cb = (int)cutoff[b];
        const float a[4] = {v.x, v.y, v.z, v.w};
#pragma unroll
        for (int k = 0; k < 4; ++k) {
            const float s = a[k];
            if (s > 0.5f && bucket_of(s) >= cb) {
                uint32_t slot = atomicAdd(&counts[b], 1u);
                if (slot < (uint32_t)CAP) {
                    uint32_t bits = __float_as_uint(s);
                    uint32_t idx = (uint32_t)(off + k);
                    cand[(long)b * CAP + slot] =
                        ((unsigned long long)bits << 32) | (uint32_t)(~idx);
                }
            }
        }
    }
}

// ---- Per-batch bitonic sort (descending), keep top 300 keys ----
__global__ void k_sort(const unsigned long long* __restrict__ cand,
                       const uint32_t* __restrict__ counts,
                       unsigned long long* __restrict__ sel) {
    __shared__ unsigned long long keys[CAP];      // 32 KB
    const int b = blockIdx.x;
    uint32_t cnt = counts[b];
    if (cnt > (uint32_t)CAP) cnt = CAP;
    for (int i = threadIdx.x; i < CAP; i += blockDim.x)
        keys[i] = (i < (int)cnt) ? cand[(long)b * CAP + i] : 0ull;
    __syncthreads();
    for (int k = 2; k <= CAP; k <<= 1) {
        for (int j = k >> 1; j > 0; j >>= 1) {
            for (int i = threadIdx.x; i < CAP; i += blockDim.x) {
                const int p = i ^ j;
                if (p > i) {
                    unsigned long long x = keys[i], y = keys[p];
                    const bool desc = ((i & k) == 0);
                    if (desc ? (x < y) : (x > y)) { keys[i] = y; keys[p] = x; }
                }
            }
            __syncthreads();
        }
    }
    for (int i = threadIdx.x; i < MAXDET; i += blockDim.x)
        sel[(long)b * MAXDET + i] = keys[i];
}

// ---- Gather outputs for each of B*300 detection slots ----
__global__ void k_gather(const float* __restrict__ boxes3D,
                         const float* __restrict__ locations,
                         const float* __restrict__ translation,
                         const float* __restrict__ rotation,
                         const float* __restrict__ confidence,
                         const unsigned long long* __restrict__ sel,
                         float* __restrict__ out) {
    const int det = blockIdx.x;                // 0 .. B*300-1
    const int b = det / MAXDET;
    const unsigned long long key = sel[det];
    const float score = __uint_as_float((uint32_t)(key >> 32));
    const bool valid = score > 0.5f;
    const uint32_t flat = ~((uint32_t)key);
    const int anchor = valid ? (int)(flat / (uint32_t)C) : 0;
    const int label  = valid ? (int)(flat % (uint32_t)C) : 0;
    const long aBase = (long)b * N + anchor;
    const float neg1 = -1.0f;

    if (threadIdx.x == 0) {
        out[OFF_SCORES + det] = valid ? score : neg1;
        out[OFF_LABELS + det] = valid ? (float)label : neg1;
        out[OFF_IDX + det]    = valid ? (float)anchor : neg1;
    }
    for (int i = threadIdx.x; i < 16; i += blockDim.x)
        out[OFF_B3 + (long)det * 16 + i] = valid ? boxes3D[aBase * 16 + i] : neg1;
    for (int i = threadIdx.x; i < 2; i += blockDim.x)
        out[OFF_LOC + (long)det * 2 + i] = valid ? locations[aBase * 2 + i] : neg1;
    for (int i = threadIdx.x; i < C * 3; i += blockDim.x)
        out[OFF_TR + (long)det * C * 3 + i] =
            valid ? translation[aBase * C * 3 + i] : neg1;
    for (int i = threadIdx.x; i < C * 6; i += blockDim.x)
        out[OFF_ROT + (long)det * C * 6 + i] =
            valid ? rotation[aBase * C * 6 + i] : neg1;
    for (int c = threadIdx.x; c < C; c += blockDim.x) {
        float s = 0.0f;
        const float* cp = confidence + (aBase * C + c) * DCONF;
#pragma unroll
        for (int k = 0; k < DCONF; ++k) s += fabsf(cp[k]);
        out[OFF_CONF + (long)det * C + c] = valid ? s : neg1;
    }
}

extern "C" void kernel_launch(void* const* d_in, const int* in_sizes, int n_in,
                              void* d_out, int out_size, void* d_ws, size_t ws_size,
                              hipStream_t stream) {
    const float* boxes3D     = (const float*)d_in[0];
    const float* cls         = (const float*)d_in[1];
    const float* locations   = (const float*)d_in[2];
    const float* translation = (const float*)d_in[3];
    const float* rotation    = (const float*)d_in[4];
    const float* confidence  = (const float*)d_in[5];
    float* out = (float*)d_out;

    uint8_t* ws = (uint8_t*)d_ws;
    uint32_t* hist   = (uint32_t*)(ws + WS_HIST);
    uint32_t* counts = (uint32_t*)(ws + WS_COUNT);
    uint32_t* cutoff = (uint32_t*)(ws + WS_CUTOFF);
    unsigned long long* cand = (unsigned long long*)(ws + WS_CAND);
    unsigned long long* sel  = (unsigned long long*)(ws + WS_SEL);

    k_init<<<64, 256, 0, stream>>>(hist, counts);

    const int totalChunks = (FLAT + CHUNK - 1) / CHUNK;       // 733
    const int blocksPerBatch = 64;
    const int chunksPerBlock = (totalChunks + blocksPerBatch - 1) / blocksPerBatch;
    k_hist<<<B * blocksPerBatch, 256, 0, stream>>>(cls, hist, chunksPerBlock,
                                                   blocksPerBatch);
    k_cutoff<<<B, 32, 0, stream>>>(hist, cutoff);
    k_collect<<<512, 256, 0, stream>>>(cls, cutoff, counts, cand);
    k_sort<<<B, 1024, 0, stream>>>(cand, counts, sel);
    k_gather<<<B * MAXDET, 256, 0, stream>>>(boxes3D, locations, translation,
                                             rotation, confidence, sel, out);
}